// KNN_21904333209873
// MI455X (gfx1250) — compile-verified
//
#include <hip/hip_runtime.h>

typedef __attribute__((ext_vector_type(2))) float v2f;
typedef __attribute__((ext_vector_type(8))) float v8f;

#define BATCH 8
#define NPTS  16384
#define NCTR  1024
#define KNN   16
#define WAVES 8
#define PTS_PER_WAVE (NPTS / WAVES)          // 2048
#define TILES_PER_WAVE (PTS_PER_WAVE / 16)   // 128
#define FLT_BIG 3.402823466e+38f

// Branchless sorted insert (ascending): carry the candidate through the list
// with single-condition compare-swaps -> v_cmp + v_cndmask chains, no exec-
// mask control flow. Guarded so the pass only runs when the candidate beats
// the current 16th-smallest (v_cmpx + s_cbranch_execz skip).
#define INSERT16(dv, iv, cd, ci)                                         \
  do {                                                                   \
    if ((cd) < dv[KNN - 1]) {                                            \
      float _c = (cd);                                                   \
      int _ci = (ci);                                                    \
      _Pragma("unroll")                                                  \
      for (int _j = 0; _j < KNN; ++_j) {                                 \
        const bool _sw = _c < dv[_j];                                    \
        const float _lo = _sw ? _c : dv[_j];                             \
        const float _hi = _sw ? dv[_j] : _c;                             \
        const int _li = _sw ? _ci : iv[_j];                              \
        const int _hiI = _sw ? iv[_j] : _ci;                             \
        dv[_j] = _lo;                                                    \
        iv[_j] = _li;                                                    \
        _c = _hi;                                                        \
        _ci = _hiI;                                                      \
      }                                                                  \
    }                                                                    \
  } while (0)

__global__ __launch_bounds__(256) void knn_topk_wmma(
    const float* __restrict__ xyz,   // (B, NPTS, 3)
    const float* __restrict__ ctr,   // (B, NCTR, 3)
    int* __restrict__ out)           // (B, KNN, NCTR)
{
  const int lane = threadIdx.x & 31;
  const int wave = threadIdx.x >> 5;
  const int b    = blockIdx.y;        // batch
  const int cbase = blockIdx.x * 16;  // 16 centers per block
  const int lc   = lane & 15;         // center column owned by this lane
  const bool hiHalf = lane >= 16;

  // ---- B operand: per-lane (-2cx,-2cy) or (-2cz, 1); C init = ||c||^2 ----
  const float* cp = ctr + ((size_t)b * NCTR + (cbase + lc)) * 3;
  const float ccx = cp[0], ccy = cp[1], ccz = cp[2];
  const float csq = ccx * ccx + ccy * ccy + ccz * ccz;
  v2f bmat;
  bmat.x = hiHalf ? (-2.0f * ccz) : (-2.0f * ccx);
  bmat.y = hiHalf ? 1.0f          : (-2.0f * ccy);

  v8f cacc;
#pragma unroll
  for (int j = 0; j < 8; ++j) cacc[j] = csq;

  // ---- per-lane running top-16 (sorted ascending) ----
  float dv[KNN];
  int   iv[KNN];
#pragma unroll
  for (int j = 0; j < KNN; ++j) { dv[j] = FLT_BIG; iv[j] = 0; }

  const int pstart = wave * PTS_PER_WAVE;
  const float* xp = xyz + ((size_t)b * NPTS + pstart + lc) * 3;

  // Software pipeline: tile t+1's coords load in flight while tile t's
  // WMMA + insert work executes, hiding L2/HBM load latency.
  float x = xp[0], y = xp[1], z = xp[2];

  for (int t = 0; t < TILES_PER_WAVE; ++t) {
    const float* xnext = xp + 16 * 3;
    // clamp prefetch on the last iteration (uniform scalar select, no OOB)
    const float* xl = (t == TILES_PER_WAVE - 1) ? xp : xnext;
    const float nx = xl[0], ny = xl[1], nz = xl[2];

    // A operand: lane<16 -> (x,y) ; lane>=16 -> (z, x^2+y^2+z^2)
    const float sq = x * x + y * y + z * z;
    v2f amat;
    amat.x = hiHalf ? z  : x;
    amat.y = hiHalf ? sq : y;

    // D = A*B + C  ==  ||x||^2 + ||c||^2 - 2 x.c  (16 pts x 16 ctrs tile)
    v8f d = __builtin_amdgcn_wmma_f32_16x16x4_f32(
        false, amat, false, bmat, (short)0, cacc, false, false);

    const int pb = pstart + t * 16 + (hiHalf ? 8 : 0);
#pragma unroll
    for (int j = 0; j < 8; ++j) {
      const float cd = d[j];
      const int   ci = pb + j;
      INSERT16(dv, iv, cd, ci);
    }

    x = nx; y = ny; z = nz;
    xp = xnext;
  }

  // ---- dump per-lane lists to LDS, merge 256 candidates per center ----
  __shared__ float sd[WAVES * 32 * KNN];   // 16 KB
  __shared__ int   si[WAVES * 32 * KNN];   // 16 KB
  const int lbase = (wave * 32 + lane) * KNN;
#pragma unroll
  for (int j = 0; j < KNN; ++j) { sd[lbase + j] = dv[j]; si[lbase + j] = iv[j]; }
  __syncthreads();

  if (threadIdx.x < 16) {
    const int c = threadIdx.x;
    float fd[KNN];
    int   fi[KNN];
#pragma unroll
    for (int j = 0; j < KNN; ++j) { fd[j] = FLT_BIG; fi[j] = 0; }

    for (int w = 0; w < WAVES; ++w) {
      for (int h = 0; h < 2; ++h) {
        const int src = (w * 32 + c + h * 16) * KNN;
        for (int j = 0; j < KNN; ++j) {          // source list is sorted:
          const float cd = sd[src + j];
          if (cd >= fd[KNN - 1]) break;          // early out
          const int ci = si[src + j];
          INSERT16(fd, fi, cd, ci);
        }
      }
    }

    // out[b][j][center] = index of j-th nearest point
    const size_t ob = (size_t)b * (KNN * NCTR) + (size_t)(cbase + c);
#pragma unroll
    for (int j = 0; j < KNN; ++j) out[ob + (size_t)j * NCTR] = fi[j];
  }
}

extern "C" void kernel_launch(void* const* d_in, const int* in_sizes, int n_in,
                              void* d_out, int out_size, void* d_ws, size_t ws_size,
                              hipStream_t stream) {
  const float* xyz = (const float*)d_in[0];  // (8, 16384, 3) f32
  const float* ctr = (const float*)d_in[1];  // (8, 1024, 3) f32
  int* out = (int*)d_out;                    // (8, 16, 1024) i32

  dim3 grid(NCTR / 16, BATCH);               // 64 x 8 blocks
  knn_topk_wmma<<<grid, 256, 0, stream>>>(xyz, ctr, out);
}